// SVDropout2D_52759378264400
// MI455X (gfx1250) — compile-verified
//
#include <hip/hip_runtime.h>

// out[i,j] = x[i,j] * (log_sigma[j] < 1.0f ? e : 1.0f)
// HBM-streaming kernel: 2.15 GB moved, ~92 us floor at 23.3 TB/s.
// B128 NT loads/stores, 8 rows per thread with a 16 KB row stride so all 8
// vmem ops share one base address + 24-bit immediate offsets (IOFFSET),
// minimizing VALU addressing work and VGPR pressure for max occupancy.

typedef float v4f __attribute__((ext_vector_type(4)));

#define THREADS 256
#define ITERS   8                       // 8 consecutive rows per thread

static constexpr long ROW4    = 4096 / 4;              // 1024 float4 per row
static constexpr long TOTAL4  = 65536L * 4096L / 4L;   // 67,108,864 float4
static constexpr long NTHREAD = TOTAL4 / ITERS;        // 8,388,608 threads
static constexpr int  BLOCKS  = (int)(NTHREAD / THREADS); // 32,768 blocks

__global__ __launch_bounds__(THREADS)
void svdropout2d_scale_kernel(const float* __restrict__ x,
                              const float* __restrict__ log_sigma,
                              float* __restrict__ out)
{
    const long g = (long)blockIdx.x * THREADS + threadIdx.x; // 0 .. NTHREAD-1

    // Each thread owns column group col4 of 8 consecutive rows.
    // Iteration stride = 1 row = 1024 float4 = 16384 bytes -> fits IOFFSET.
    const int  col4  = (int)(g & 1023);
    const long base4 = (g >> 10) * (ITERS * ROW4) + col4;

    // log_sigma is 16 KB total: regular-temporal load, stays hot in WGP$/L2.
    const v4f ls = *reinterpret_cast<const v4f*>(log_sigma + (long)col4 * 4);

    const float E = 2.71828182845904523536f; // exp(1.0f)
    v4f scale;
    scale.x = (ls.x < 1.0f) ? E : 1.0f;
    scale.y = (ls.y < 1.0f) ? E : 1.0f;
    scale.z = (ls.z < 1.0f) ? E : 1.0f;
    scale.w = (ls.w < 1.0f) ? E : 1.0f;

    const v4f* __restrict__ xv = reinterpret_cast<const v4f*>(x) + base4;
    v4f*       __restrict__ ov = reinterpret_cast<v4f*>(out) + base4;

    // 8 clause'd B128 NT loads off one base pointer (immediate offsets),
    // then 8 multiply + B128 NT stores. x/out are streamed once: NT keeps
    // them from rinsing the 192 MB L2.
    v4f v[ITERS];
#pragma unroll
    for (int k = 0; k < ITERS; ++k)
        v[k] = __builtin_nontemporal_load(xv + k * ROW4);

#pragma unroll
    for (int k = 0; k < ITERS; ++k)
        __builtin_nontemporal_store(v[k] * scale, ov + k * ROW4);
}

extern "C" void kernel_launch(void* const* d_in, const int* in_sizes, int n_in,
                              void* d_out, int out_size, void* d_ws, size_t ws_size,
                              hipStream_t stream)
{
    const float* x         = (const float*)d_in[0]; // [65536, 4096] f32
    const float* log_sigma = (const float*)d_in[1]; // [4096] f32
    float*       out       = (float*)d_out;         // [65536, 4096] f32

    svdropout2d_scale_kernel<<<BLOCKS, THREADS, 0, stream>>>(x, log_sigma, out);
}